// HungarianMatcher_56169582297541
// MI455X (gfx1250) — compile-verified
//
#include <hip/hip_runtime.h>
#include <stdint.h>

// ---------------------------------------------------------------------------
// HungarianMatcher for MI455X (gfx1250, wave32).
//   bs=32, Q=256, T=128, NC=2048, bs*T=4096
//   out = [ C : f32[32*256*4096] | rows : f32[32*128] | cols : f32[32*128] ]
// Phase 1 (bandwidth-bound): fused softmax + index-gather; tgt_ids staged
//   into LDS with the Tensor Data Mover (async DMA, TENSORcnt) overlapped
//   with the softmax math. Also emits a transposed per-batch cost block to
//   scratch for phase 2.
// Phase 2 (latency-bound, 32 independent instances): scipy-exact
//   Jonker-Volgenant shortest augmenting path in float64, one workgroup per
//   batch instance.
// ---------------------------------------------------------------------------

namespace {
constexpr int kBS = 32;
constexpr int kQ  = 256;
constexpr int kT  = 128;
constexpr int kNC = 2048;
constexpr int kBT = kBS * kT;                        // 4096
constexpr size_t kCElems = (size_t)kBS * kQ * kBT;   // 33,554,432
}

typedef unsigned int v4u __attribute__((ext_vector_type(4)));
typedef int          v8i __attribute__((ext_vector_type(8)));
typedef int          v4i __attribute__((ext_vector_type(4)));

__device__ __forceinline__ float block_reduce_max(float v, float* sred) {
  const int t = threadIdx.x;
  sred[t] = v;
  __syncthreads();
  for (int s = 128; s > 0; s >>= 1) {
    if (t < s) sred[t] = fmaxf(sred[t], sred[t + s]);
    __syncthreads();
  }
  float r = sred[0];
  __syncthreads();
  return r;
}

__device__ __forceinline__ float block_reduce_sum(float v, float* sred) {
  const int t = threadIdx.x;
  sred[t] = v;
  __syncthreads();
  for (int s = 128; s > 0; s >>= 1) {
    if (t < s) sred[t] += sred[t + s];
    __syncthreads();
  }
  float r = sred[0];
  __syncthreads();
  return r;
}

// ---------------------------------------------------------------------------
// Kernel 1: one block per (b, q) row.
// ---------------------------------------------------------------------------
__global__ __launch_bounds__(256) void softmax_gather_kernel(
    const float* __restrict__ logits,          // [32,256,2048]
    const int* __restrict__ tgt,               // [32,128] flattened ids
    const unsigned char* __restrict__ amask,   // [32,256] bool
    const int* __restrict__ ubp,               // [1] use_background
    float* __restrict__ Cout,                  // [32*256, 4096]
    float* __restrict__ wsT)                   // [32][128][256] transposed blocks
{
  __shared__ float s_p[kNC];        // 8 KB  : softmax probabilities
  __shared__ int   s_tids[kBT];     // 16 KB : gather index table (TDM target)
  __shared__ float sred[256];

  const int t   = threadIdx.x;
  const int row = blockIdx.x;       // b*256 + q
  const int b   = row >> 8;
  const int q   = row & 255;

  // --- Tensor Data Mover: async 16 KB copy of tgt_ids -> LDS --------------
  // D# group0/group1 packed per CDNA5 ISA:
  //   1D tile: data_size=4B, tensor_dim0=tile_dim0=4096, tensor_dim1=tile_dim1=1
  if (t == 0) {
    uint64_t ga = (uint64_t)(uintptr_t)tgt;
    uint32_t la = (uint32_t)(uintptr_t)(&s_tids[0]);
    v4u g0 = { 1u,                                              // count=1
               la,                                              // lds_addr
               (uint32_t)ga,                                    // global addr lo
               (uint32_t)((ga >> 32) & 0x01FFFFFFu) | (2u << 30) }; // hi | type=2
    v8i g1 = { (int)(2u << 16),                                 // data_size=4B
               (int)(((unsigned)kBT & 0xFFFFu) << 16),          // tensor_dim0 lo16
               (int)(1u << 16),                                 // dim0 hi=0 | tensor_dim1=1
               (int)(((unsigned)kBT & 0xFFFFu) << 16),          // dim1 hi=0 | tile_dim0
               1,                                               // tile_dim1=1, tile_dim2=0
               kBT,                                             // tensor_dim0_stride lo32
               0, 0 };
    v4i z4 = {0, 0, 0, 0};
    v8i z8 = {0, 0, 0, 0, 0, 0, 0, 0};
    __builtin_amdgcn_tensor_load_to_lds(g0, g1, z4, z4, z8, 0);
  }

  // --- f32 softmax of this row (overlaps the TDM transfer) ----------------
  const float* lrow = logits + (size_t)row * kNC;
  float4 x0 = *(const float4*)(lrow + t * 4);          // [0, 1024)
  float4 x1 = *(const float4*)(lrow + 1024 + t * 4);   // [1024, 2048)

  float m = fmaxf(fmaxf(fmaxf(x0.x, x0.y), fmaxf(x0.z, x0.w)),
                  fmaxf(fmaxf(x1.x, x1.y), fmaxf(x1.z, x1.w)));
  m = block_reduce_max(m, sred);

  float4 e0, e1;
  e0.x = expf(x0.x - m); e0.y = expf(x0.y - m);
  e0.z = expf(x0.z - m); e0.w = expf(x0.w - m);
  e1.x = expf(x1.x - m); e1.y = expf(x1.y - m);
  e1.z = expf(x1.z - m); e1.w = expf(x1.w - m);

  float s = (e0.x + e0.y) + (e0.z + e0.w) + (e1.x + e1.y) + (e1.z + e1.w);
  s = block_reduce_sum(s, sred);
  const float inv = 1.0f / s;

  *(float4*)(s_p + t * 4)        = make_float4(e0.x*inv, e0.y*inv, e0.z*inv, e0.w*inv);
  *(float4*)(s_p + 1024 + t * 4) = make_float4(e1.x*inv, e1.y*inv, e1.z*inv, e1.w*inv);

  const int ub = ubp[0];
  const float scale = (ub != 0 || amask[row] != 0) ? 1.0f : 0.0f;

  // wait for TDM completion (issuing wave), then make LDS visible to all
  if (t == 0) __builtin_amdgcn_s_wait_tensorcnt(0);
  __syncthreads();

  // --- gather: C[row, j] = -(p[tgt[j]] * mask) ----------------------------
  float* crow = Cout + (size_t)row * kBT;
#pragma unroll
  for (int jj = 0; jj < kBT / 256; ++jj) {
    const int j  = jj * 256 + t;
    const int id = s_tids[j];
    const float vv = -(s_p[id] * scale);
    crow[j] = vv;
    // own-batch slice, stored transposed for the Hungarian kernel
    if ((j >> 7) == b) {
      wsT[((size_t)b * kT + (j & 127)) * kQ + q] = vv;
    }
  }
}

// ---------------------------------------------------------------------------
// Kernel 2: scipy-style Jonker-Volgenant on [nr=128, nc=256] (transposed
// block, since Q > T). One workgroup per batch instance, f64 duals in LDS.
// ---------------------------------------------------------------------------
__global__ __launch_bounds__(256) void hungarian_kernel(
    const float* __restrict__ wsT,   // [32][128][256]
    float* __restrict__ rows_out,    // [32*128]
    float* __restrict__ cols_out)    // [32*128]
{
  __shared__ double u[kT], v[kQ], d[kQ];
  __shared__ double s_val[kQ];
  __shared__ int    s_idx[kQ];
  __shared__ int    path[kQ], row4col[kQ], col4row[kT];
  __shared__ unsigned char SR[kT], SC[kQ];
  __shared__ int    s_i, s_sink;
  __shared__ double s_min;

  const int t = threadIdx.x;
  const int b = blockIdx.x;
  const float* cost = wsT + (size_t)b * kT * kQ;

  if (t < kT) { u[t] = 0.0; col4row[t] = -1; }
  v[t] = 0.0; row4col[t] = -1;
  __syncthreads();

  const double INF = __builtin_inf();

  for (int cur = 0; cur < kT; ++cur) {
    if (t < kT) SR[t] = 0;
    SC[t] = 0;
    d[t]  = INF;
    if (t == 0) { s_i = cur; s_sink = -1; s_min = 0.0; }
    __syncthreads();

    for (;;) {
      const int    i  = s_i;
      const double mv = s_min;
      const double ui = u[i];
      if (t == 0) SR[i] = 1;
      if (!SC[t]) {
        double r = mv + (double)cost[i * kQ + t] - ui - v[t];
        if (r < d[t]) { d[t] = r; path[t] = i; }
      }
      s_val[t] = SC[t] ? INF : d[t];
      s_idx[t] = t;
      __syncthreads();

      // argmin with numpy first-occurrence tie-break (smallest index)
      for (int s = 128; s > 0; s >>= 1) {
        if (t < s) {
          const double v2 = s_val[t + s];
          const int    i2 = s_idx[t + s];
          if (v2 < s_val[t] || (v2 == s_val[t] && i2 < s_idx[t])) {
            s_val[t] = v2; s_idx[t] = i2;
          }
        }
        __syncthreads();
      }

      if (t == 0) {
        const int j = s_idx[0];
        s_min = s_val[0];
        SC[j] = 1;
        if (row4col[j] < 0) s_sink = j;
        else                s_i    = row4col[j];
      }
      __syncthreads();
      if (s_sink >= 0) break;
    }

    // dual updates BEFORE augmentation (scipy _lsap_body order)
    if (t == 0) u[cur] += s_min;
    if (t < kT && t != cur && SR[t]) u[t] += s_min - d[col4row[t]];
    if (SC[t]) v[t] -= s_min - d[t];
    __syncthreads();

    // augment along shortest path (serial, lane 0)
    if (t == 0) {
      int j = s_sink;
      for (;;) {
        const int ii = path[j];
        row4col[j] = ii;
        const int jn = col4row[ii];
        col4row[ii] = j;
        j = jn;
        if (ii == cur) break;
      }
    }
    __syncthreads();
  }

  // transposed-return semantics: order = argsort(col4row);
  // rows = col4row[order] (sorted query ids), cols = order (target ids)
  if (t < kT) {
    const int qq = col4row[t];
    int rank = 0;
    for (int k = 0; k < kT; ++k) rank += (col4row[k] < qq) ? 1 : 0;
    rows_out[b * kT + rank] = (float)qq;
    cols_out[b * kT + rank] = (float)t;
  }
}

// ---------------------------------------------------------------------------
extern "C" void kernel_launch(void* const* d_in, const int* in_sizes, int n_in,
                              void* d_out, int out_size, void* d_ws, size_t ws_size,
                              hipStream_t stream) {
  (void)in_sizes; (void)n_in; (void)out_size; (void)ws_size;

  const float*         logits = (const float*)d_in[0];
  const int*           tgt    = (const int*)d_in[1];
  const unsigned char* amask  = (const unsigned char*)d_in[2];
  const int*           ubp    = (const int*)d_in[3];

  float* out  = (float*)d_out;
  float* wsT  = (float*)d_ws;                 // 32*128*256 f32 = 4 MB scratch
  float* rows = out + kCElems;
  float* cols = rows + (size_t)kBS * kT;

  softmax_gather_kernel<<<kBS * kQ, 256, 0, stream>>>(logits, tgt, amask, ubp,
                                                      out, wsT);
  hungarian_kernel<<<kBS, 256, 0, stream>>>(wsT, rows, cols);
}